// Coattention_68401649156852
// MI455X (gfx1250) — compile-verified
//
#include <hip/hip_runtime.h>
#include <hip/hip_bf16.h>

#define BATCH 8
#define CL    2048
#define QL    512
#define DIMN  512

typedef __attribute__((ext_vector_type(2))) float v2f;
typedef __attribute__((ext_vector_type(8))) float v8f;

// f32 matrix core: D(16x16) = A(16x4) * B(4x16) + C
__device__ __forceinline__ v8f wmma4(v2f a, v2f b, v8f c) {
  return __builtin_amdgcn_wmma_f32_16x16x4_f32(
      /*neg_a=*/false, a, /*neg_b=*/false, b,
      /*c_mod=*/(short)0, c, /*reuse_a=*/false, /*reuse_b=*/false);
}

__device__ __forceinline__ float waveSum(float v) {
  #pragma unroll
  for (int off = 16; off > 0; off >>= 1) v += __shfl_xor(v, off, 32);
  return v;
}
__device__ __forceinline__ float waveMax(float v) {
  #pragma unroll
  for (int off = 16; off > 0; off >>= 1) v = fmaxf(v, __shfl_xor(v, off, 32));
  return v;
}
__device__ __forceinline__ void atomicMaxF(float* addr, float v) {
  unsigned int* ua = (unsigned int*)addr;
  unsigned int old = *ua;
  while (__uint_as_float(old) < v) {
    unsigned int assumed = old;
    old = atomicCAS(ua, assumed, __float_as_uint(v));
    if (old == assumed) break;
  }
}

// ---------------- k1: Qp[b,j,e] = sum_d Q[j,b,d]*Wq[d,e] + bq[e] ----------------
// 1024 waves: b(8) x jm(16) x en(8); wave tile = 32(j) x 64(e); B fetch shared by 2 WMMAs
__global__ __launch_bounds__(256) void k_qp(const float* __restrict__ Q,
                                            const float* __restrict__ Wq,
                                            const float* __restrict__ bq,
                                            float* __restrict__ Qp) {
  int wid = (int)((blockIdx.x * blockDim.x + threadIdx.x) >> 5);
  const int lane = threadIdx.x & 31, row = lane & 15, kh = lane >> 4;
  const int en = wid & 7;  wid >>= 3;
  const int jm = wid & 15; wid >>= 4;
  const int b  = wid;
  const int j0 = jm * 32, e0 = en * 64;

  const float* qrow0 = Q + ((size_t)(j0 + row) * BATCH + b) * DIMN;
  const float* qrow1 = Q + ((size_t)(j0 + 16 + row) * BATCH + b) * DIMN;
  v8f acc[2][4] = {};
  for (int k = 0; k < DIMN; k += 4) {
    const int ka = k + 2 * kh;
    v2f a0; a0.x = qrow0[ka]; a0.y = qrow0[ka + 1];
    v2f a1; a1.x = qrow1[ka]; a1.y = qrow1[ka + 1];
    #pragma unroll
    for (int n = 0; n < 4; ++n) {
      const int e = e0 + 16 * n + row;
      v2f bb; bb.x = Wq[(size_t)ka * DIMN + e]; bb.y = Wq[(size_t)(ka + 1) * DIMN + e];
      acc[0][n] = wmma4(a0, bb, acc[0][n]);
      acc[1][n] = wmma4(a1, bb, acc[1][n]);
    }
  }
  #pragma unroll
  for (int n = 0; n < 4; ++n) {
    const int e = e0 + 16 * n + row;
    const float bqe = bq[e];
    #pragma unroll
    for (int h = 0; h < 2; ++h)
      #pragma unroll
      for (int vg = 0; vg < 8; ++vg) {
        const int j = j0 + 16 * h + vg + 8 * kh;
        Qp[((size_t)b * QL + j) * DIMN + e] = acc[h][n][vg] + bqe;
      }
  }
}

// ---------------- k2a: r1[b,i] = C[i,b,:].w1 ----------------
__global__ __launch_bounds__(256) void k_r1(const float* __restrict__ C,
                                            const float* __restrict__ w1,
                                            float* __restrict__ r1) {
  const int wid = (int)((blockIdx.x * blockDim.x + threadIdx.x) >> 5);
  const int lane = threadIdx.x & 31;
  const int b = wid / CL, i = wid % CL;
  const float* crow = C + ((size_t)i * BATCH + b) * DIMN;
  float s = 0.f;
  for (int d = lane; d < DIMN; d += 32) s += crow[d] * w1[d];
  s = waveSum(s);
  if (lane == 0) r1[b * CL + i] = s;
}

// ---------------- k2b: r2[b,j] = Qp[b,j,:].w2 ----------------
__global__ __launch_bounds__(256) void k_r2(const float* __restrict__ Qp,
                                            const float* __restrict__ w2,
                                            float* __restrict__ r2) {
  const int wid = (int)((blockIdx.x * blockDim.x + threadIdx.x) >> 5);
  const int lane = threadIdx.x & 31;
  const int b = wid / QL, j = wid % QL;
  const float* qpr = Qp + ((size_t)b * QL + j) * DIMN;
  float s = 0.f;
  for (int d = lane; d < DIMN; d += 32) s += qpr[d] * w2[d];
  s = waveSum(s);
  if (lane == 0) r2[b * QL + j] = s;
}

// ---------------- k3: S[b,i,j] = (C*w3).Qp + r1 + r2 + b ----------------
// 4096 waves: b(8) x im(64) x jn(8); tile 32(i) x 64(j), K=512 over d
__global__ __launch_bounds__(256) void k_score(const float* __restrict__ C,
                                               const float* __restrict__ Qp,
                                               const float* __restrict__ w3,
                                               const float* __restrict__ r1,
                                               const float* __restrict__ r2,
                                               const float* __restrict__ bscal,
                                               float* __restrict__ S) {
  int wid = (int)((blockIdx.x * blockDim.x + threadIdx.x) >> 5);
  const int lane = threadIdx.x & 31, row = lane & 15, kh = lane >> 4;
  const int jn = wid & 7;  wid >>= 3;
  const int im = wid & 63; wid >>= 6;
  const int b  = wid;
  const int i0 = im * 32, j0 = jn * 64;

  const float* crow0 = C + ((size_t)(i0 + row) * BATCH + b) * DIMN;
  const float* crow1 = C + ((size_t)(i0 + 16 + row) * BATCH + b) * DIMN;
  v8f acc[2][4] = {};
  for (int k = 0; k < DIMN; k += 4) {
    const int ka = k + 2 * kh;
    const float w3a = w3[ka], w3b = w3[ka + 1];
    v2f a0; a0.x = crow0[ka] * w3a; a0.y = crow0[ka + 1] * w3b;
    v2f a1; a1.x = crow1[ka] * w3a; a1.y = crow1[ka + 1] * w3b;
    #pragma unroll
    for (int n = 0; n < 4; ++n) {
      const int j = j0 + 16 * n + row;
      const float* qpr = Qp + ((size_t)b * QL + j) * DIMN;
      v2f bb; bb.x = qpr[ka]; bb.y = qpr[ka + 1];
      acc[0][n] = wmma4(a0, bb, acc[0][n]);
      acc[1][n] = wmma4(a1, bb, acc[1][n]);
    }
  }
  const float bc = bscal[0];
  float r1v[2][8];
  #pragma unroll
  for (int h = 0; h < 2; ++h)
    #pragma unroll
    for (int vg = 0; vg < 8; ++vg)
      r1v[h][vg] = r1[b * CL + i0 + 16 * h + vg + 8 * kh];
  #pragma unroll
  for (int n = 0; n < 4; ++n) {
    const int j = j0 + 16 * n + row;
    const float r2v = r2[b * QL + j];
    #pragma unroll
    for (int h = 0; h < 2; ++h)
      #pragma unroll
      for (int vg = 0; vg < 8; ++vg) {
        const int i = i0 + 16 * h + vg + 8 * kh;
        S[((size_t)b * CL + i) * QL + j] = acc[h][n][vg] + r1v[h][vg] + r2v + bc;
      }
  }
}

// ---------------- k4: row softmax stats over j (len 512), one wave per (b,i) ----------------
__global__ __launch_bounds__(256) void k_rowstats(const float* __restrict__ S,
                                                  const int* __restrict__ Qm,
                                                  float* __restrict__ rowmax,
                                                  float* __restrict__ rowinv) {
  const int wid = (int)((blockIdx.x * blockDim.x + threadIdx.x) >> 5);
  const int lane = threadIdx.x & 31;
  const int b = wid / CL, i = wid % CL;
  const float* srow = S + ((size_t)b * CL + i) * QL;
  const float NEG = -__builtin_inff();
  float vals[16];
  float mx = NEG;
  #pragma unroll
  for (int t = 0; t < 16; ++t) {
    const int j = lane + 32 * t;
    const float v = (Qm[j * BATCH + b] != 0) ? srow[j] : NEG;
    vals[t] = v;
    mx = fmaxf(mx, v);
  }
  mx = waveMax(mx);
  float s = 0.f;
  #pragma unroll
  for (int t = 0; t < 16; ++t)
    if (vals[t] > NEG) s += __expf(vals[t] - mx);
  s = waveSum(s);
  if (lane == 0) {
    rowmax[b * CL + i] = (mx > NEG) ? mx : 0.f;
    rowinv[b * CL + i] = (s > 0.f) ? 1.f / s : 0.f;
  }
}

// ---------------- col stats (over i, len 2048): init, max, sum ----------------
__global__ __launch_bounds__(256) void k_initcol(float* __restrict__ colmax,
                                                 float* __restrict__ colsum) {
  const int t = blockIdx.x * 256 + threadIdx.x;
  if (t < BATCH * QL) { colmax[t] = -__builtin_inff(); colsum[t] = 0.f; }
}

// blocks: b(8) x jb(2) x ic(8) = 128; thread j = jb*256+tid, i-chunk of 256
__global__ __launch_bounds__(256) void k_colmax(const float* __restrict__ S,
                                                const int* __restrict__ Cm,
                                                float* __restrict__ colmax) {
  int bx = blockIdx.x;
  const int ic = bx & 7; bx >>= 3;
  const int jb = bx & 1; bx >>= 1;
  const int b  = bx;
  const int j = jb * 256 + threadIdx.x;
  float mx = -__builtin_inff();
  const int ib = ic * 256;
  for (int ii = 0; ii < 256; ++ii) {
    const int i = ib + ii;
    if (Cm[i * BATCH + b] != 0) mx = fmaxf(mx, S[((size_t)b * CL + i) * QL + j]);
  }
  atomicMaxF(&colmax[b * QL + j], mx);
}

__global__ __launch_bounds__(256) void k_colsum(const float* __restrict__ S,
                                                const int* __restrict__ Cm,
                                                const float* __restrict__ colmax,
                                                float* __restrict__ colsum) {
  int bx = blockIdx.x;
  const int ic = bx & 7; bx >>= 3;
  const int jb = bx & 1; bx >>= 1;
  const int b  = bx;
  const int j = jb * 256 + threadIdx.x;
  const float cm = colmax[b * QL + j];
  if (!(cm > -__builtin_inff())) return;  // fully masked column
  float s = 0.f;
  const int ib = ic * 256;
  for (int ii = 0; ii < 256; ++ii) {
    const int i = ib + ii;
    if (Cm[i * BATCH + b] != 0) s += __expf(S[((size_t)b * CL + i) * QL + j] - cm);
  }
  atomicAdd(&colsum[b * QL + j], s);
}

// ---------------- k6: T[b,j,e] = sum_i S_col[b,i,j] * C[i,b,e] ----------------
// 1024 waves: b(8) x jm(16) x en(8); tile 32(j) x 64(e), K=2048 over i; exp on A load
__global__ __launch_bounds__(256) void k_T(const float* __restrict__ S,
                                           const float* __restrict__ C,
                                           const int* __restrict__ Cm,
                                           const float* __restrict__ colmax,
                                           const float* __restrict__ colsum,
                                           float* __restrict__ T) {
  int wid = (int)((blockIdx.x * blockDim.x + threadIdx.x) >> 5);
  const int lane = threadIdx.x & 31, row = lane & 15, kh = lane >> 4;
  const int en = wid & 7;  wid >>= 3;
  const int jm = wid & 15; wid >>= 4;
  const int b  = wid;
  const int j0 = jm * 32, e0 = en * 64;

  const int jA0 = j0 + row, jA1 = j0 + 16 + row;
  float cm0 = colmax[b * QL + jA0], cm1 = colmax[b * QL + jA1];
  const float cs0 = colsum[b * QL + jA0], cs1 = colsum[b * QL + jA1];
  const float ci0 = (cs0 > 0.f) ? 1.f / cs0 : 0.f;
  const float ci1 = (cs1 > 0.f) ? 1.f / cs1 : 0.f;
  if (!(cm0 > -__builtin_inff())) cm0 = 0.f;  // keep exp finite; ci==0 zeroes it
  if (!(cm1 > -__builtin_inff())) cm1 = 0.f;

  v8f acc[2][4] = {};
  for (int k = 0; k < CL; k += 4) {
    const int ka = k + 2 * kh;
    const float m0 = (Cm[ka * BATCH + b] != 0) ? 1.f : 0.f;
    const float m1 = (Cm[(ka + 1) * BATCH + b] != 0) ? 1.f : 0.f;
    const float* s0 = S + ((size_t)b * CL + ka) * QL;
    const float* s1 = S + ((size_t)b * CL + ka + 1) * QL;
    v2f a0, a1;
    a0.x = m0 * __expf(s0[jA0] - cm0) * ci0;
    a0.y = m1 * __expf(s1[jA0] - cm0) * ci0;
    a1.x = m0 * __expf(s0[jA1] - cm1) * ci1;
    a1.y = m1 * __expf(s1[jA1] - cm1) * ci1;
    #pragma unroll
    for (int n = 0; n < 4; ++n) {
      const int e = e0 + 16 * n + row;
      v2f bb;
      bb.x = C[((size_t)ka * BATCH + b) * DIMN + e];
      bb.y = C[((size_t)(ka + 1) * BATCH + b) * DIMN + e];
      acc[0][n] = wmma4(a0, bb, acc[0][n]);
      acc[1][n] = wmma4(a1, bb, acc[1][n]);
    }
  }
  #pragma unroll
  for (int n = 0; n < 4; ++n) {
    const int e = e0 + 16 * n + row;
    #pragma unroll
    for (int h = 0; h < 2; ++h)
      #pragma unroll
      for (int vg = 0; vg < 8; ++vg) {
        const int j = j0 + 16 * h + vg + 8 * kh;
        T[((size_t)b * QL + j) * DIMN + e] = acc[h][n][vg];
      }
  }
}

// ---------------- k7: fused A = S_row@Qp, Bm = S_row@T, write concat(C, A, C*A, C*Bm) ----------------
// 4096 waves: b(8) x im(64) x en(8); tile 32(i) x 64(e) for BOTH GEMMs (shared A and B operands)
__global__ __launch_bounds__(256) void k_out(const float* __restrict__ S,
                                             const float* __restrict__ C,
                                             const float* __restrict__ Qp,
                                             const float* __restrict__ T,
                                             const int* __restrict__ Qm,
                                             const float* __restrict__ rowmax,
                                             const float* __restrict__ rowinv,
                                             float* __restrict__ out) {
  int wid = (int)((blockIdx.x * blockDim.x + threadIdx.x) >> 5);
  const int lane = threadIdx.x & 31, row = lane & 15, kh = lane >> 4;
  const int en = wid & 7;  wid >>= 3;
  const int im = wid & 63; wid >>= 6;
  const int b  = wid;
  const int i0 = im * 32, e0 = en * 64;

  const int iA0 = i0 + row, iA1 = i0 + 16 + row;
  const float rm0 = rowmax[b * CL + iA0], ri0 = rowinv[b * CL + iA0];
  const float rm1 = rowmax[b * CL + iA1], ri1 = rowinv[b * CL + iA1];
  const float* srow0 = S + ((size_t)b * CL + iA0) * QL;
  const float* srow1 = S + ((size_t)b * CL + iA1) * QL;

  v8f accA[2][4] = {}, accB[2][4] = {};
  for (int k = 0; k < QL; k += 4) {
    const int ka = k + 2 * kh;
    const float m0 = (Qm[ka * BATCH + b] != 0) ? 1.f : 0.f;
    const float m1 = (Qm[(ka + 1) * BATCH + b] != 0) ? 1.f : 0.f;
    v2f a0, a1;
    a0.x = m0 * __expf(srow0[ka] - rm0) * ri0;
    a0.y = m1 * __expf(srow0[ka + 1] - rm0) * ri0;
    a1.x = m0 * __expf(srow1[ka] - rm1) * ri1;
    a1.y = m1 * __expf(srow1[ka + 1] - rm1) * ri1;
    #pragma unroll
    for (int n = 0; n < 4; ++n) {
      const int e = e0 + 16 * n + row;
      v2f bq; bq.x = Qp[((size_t)b * QL + ka) * DIMN + e];
              bq.y = Qp[((size_t)b * QL + ka + 1) * DIMN + e];
      accA[0][n] = wmma4(a0, bq, accA[0][n]);
      accA[1][n] = wmma4(a1, bq, accA[1][n]);
      v2f bt; bt.x = T[((size_t)b * QL + ka) * DIMN + e];
              bt.y = T[((size_t)b * QL + ka + 1) * DIMN + e];
      accB[0][n] = wmma4(a0, bt, accB[0][n]);
      accB[1][n] = wmma4(a1, bt, accB[1][n]);
    }
  }
  #pragma unroll
  for (int n = 0; n < 4; ++n) {
    const int e = e0 + 16 * n + row;
    #pragma unroll
    for (int h = 0; h < 2; ++h)
      #pragma unroll
      for (int vg = 0; vg < 8; ++vg) {
        const int i = i0 + 16 * h + vg + 8 * kh;
        const float c  = C[((size_t)i * BATCH + b) * DIMN + e];
        const float av = accA[h][n][vg];
        const float bv = accB[h][n][vg];
        float* ob = out + ((size_t)i * BATCH + b) * (size_t)(4 * DIMN);
        // streaming, write-once output: non-temporal stores keep S/Qp/T in L2
        __builtin_nontemporal_store(c,      ob + e);
        __builtin_nontemporal_store(av,     ob + DIMN + e);
        __builtin_nontemporal_store(c * av, ob + 2 * DIMN + e);
        __builtin_nontemporal_store(c * bv, ob + 3 * DIMN + e);
      }
  }
}

extern "C" void kernel_launch(void* const* d_in, const int* in_sizes, int n_in,
                              void* d_out, int out_size, void* d_ws, size_t ws_size,
                              hipStream_t stream) {
  const float* C  = (const float*)d_in[0];
  const float* Q  = (const float*)d_in[1];
  const int*   Cm = (const int*)d_in[2];
  const int*   Qm = (const int*)d_in[3];
  const float* Wq = (const float*)d_in[4];
  const float* bq = (const float*)d_in[5];
  const float* w1 = (const float*)d_in[6];
  const float* w2 = (const float*)d_in[7];
  const float* w3 = (const float*)d_in[8];
  const float* bs = (const float*)d_in[9];
  float* out = (float*)d_out;

  float* ws = (float*)d_ws;
  float* Qp     = ws;                              // 8*512*512   = 2,097,152
  float* S      = Qp + (size_t)BATCH * QL * DIMN;  // 8*2048*512  = 8,388,608
  float* T      = S  + (size_t)BATCH * CL * QL;    // 8*512*512   = 2,097,152
  float* r1     = T  + (size_t)BATCH * QL * DIMN;  // 8*2048
  float* r2     = r1 + BATCH * CL;                 // 8*512
  float* rowmax = r2 + BATCH * QL;                 // 8*2048
  float* rowinv = rowmax + BATCH * CL;             // 8*2048
  float* colmax = rowinv + BATCH * CL;             // 8*512
  float* colsum = colmax + BATCH * QL;             // 8*512
  // total ~50.6 MB of f32 workspace

  k_qp      <<<128,  256, 0, stream>>>(Q, Wq, bq, Qp);
  k_r1      <<<2048, 256, 0, stream>>>(C, w1, r1);
  k_r2      <<<512,  256, 0, stream>>>(Qp, w2, r2);
  k_score   <<<512,  256, 0, stream>>>(C, Qp, w3, r1, r2, bs, S);
  k_rowstats<<<2048, 256, 0, stream>>>(S, Qm, rowmax, rowinv);
  k_initcol <<<16,   256, 0, stream>>>(colmax, colsum);
  k_colmax  <<<128,  256, 0, stream>>>(S, Cm, colmax);
  k_colsum  <<<128,  256, 0, stream>>>(S, Cm, colmax, colsum);
  k_T       <<<128,  256, 0, stream>>>(S, C, Cm, colmax, colsum, T);
  k_out     <<<512,  256, 0, stream>>>(S, C, Qp, T, Qm, rowmax, rowinv, out);
}